// GATEncoder_15771119911600
// MI455X (gfx1250) — compile-verified
//
#include <hip/hip_runtime.h>
#include <math.h>

#define N_NODES 65536
#define N_EDGES 1048576
#define DCH 64
#define NEG_SLOPE 0.2f

typedef __attribute__((ext_vector_type(2))) float v2f;
typedef __attribute__((ext_vector_type(8))) float v8f;

// ---------------------------------------------------------------------------
// Small prep: vdst[k] = sum_c W_dst[k,c] * a_dst[c];  cbias[c] = b[c] + bl[c]
// ---------------------------------------------------------------------------
__global__ void prep_kernel(const float* __restrict__ Wdst,
                            const float* __restrict__ adst,
                            const float* __restrict__ b,
                            const float* __restrict__ bl,
                            float* __restrict__ vdst,
                            float* __restrict__ cbias) {
    int k = threadIdx.x;  // 64 threads
    float s = 0.f;
    for (int c = 0; c < DCH; ++c) s += Wdst[k * DCH + c] * adst[c];
    vdst[k] = s;
    cbias[k] = b[k] + bl[k];
}

__global__ void init_nodes_kernel(float* __restrict__ m, float* __restrict__ denom) {
    int n = blockIdx.x * blockDim.x + threadIdx.x;
    if (n < N_NODES) { m[n] = -INFINITY; denom[n] = 0.f; }
}

// ---------------------------------------------------------------------------
// Dual GEMM via V_WMMA_F32_16X16X4_F32:
//   XS  = X @ Wa                      (waves 0..3)
//   ACC = X @ Wb + cbias              (waves 4..7)   -- output accumulator seed
// One block handles 16 rows of X. 8 waves = 4 col-tiles x 2 weight matrices.
// ---------------------------------------------------------------------------
__global__ void __launch_bounds__(256)
gemm_dual_kernel(const float* __restrict__ X,
                 const float* __restrict__ Wa,
                 const float* __restrict__ Wb,
                 const float* __restrict__ cbias,
                 float* __restrict__ XS,
                 float* __restrict__ ACC) {
    __shared__ float sX[16 * 65];      // padded stride 65 -> conflict-free A reads
    __shared__ float sWa[64 * 64];
    __shared__ float sWb[64 * 64];

    const int tid = threadIdx.x;
    const size_t rowbase = (size_t)blockIdx.x * 16;

    // Load 16x64 X tile (scalar, coalesced) into padded LDS
    for (int i = 0; i < 4; ++i) {
        int idx = tid + 256 * i;                 // 0..1023
        sX[(idx >> 6) * 65 + (idx & 63)] = X[rowbase * DCH + idx];
    }
    // Load both 64x64 weight matrices (float4, coalesced)
    for (int i = 0; i < 4; ++i) {
        ((float4*)sWa)[tid + 256 * i] = ((const float4*)Wa)[tid + 256 * i];
        ((float4*)sWb)[tid + 256 * i] = ((const float4*)Wb)[tid + 256 * i];
    }
    __syncthreads();

    const int wave = tid >> 5;          // 0..7
    const int lane = tid & 31;
    const float* Wsh = (wave < 4) ? sWa : sWb;
    const int coltile = (wave & 3) * 16;
    const int mrow  = lane & 15;
    const int khalf = lane >> 4;        // 0 -> K{0,1}, 1 -> K{2,3}
    const int ncol  = coltile + (lane & 15);

    v8f acc = {};
    for (int ks = 0; ks < 16; ++ks) {
        const int k0 = ks * 4 + khalf * 2;
        v2f a, b;
        a.x = sX[mrow * 65 + k0];
        a.y = sX[mrow * 65 + k0 + 1];
        b.x = Wsh[k0 * 64 + ncol];
        b.y = Wsh[(k0 + 1) * 64 + ncol];
        acc = __builtin_amdgcn_wmma_f32_16x16x4_f32(
            false, a, false, b, (short)0, acc, false, false);
    }

    // D layout: vgpr i -> M = khalf*8 + i, N = ncol
    const int mbase = khalf * 8;
    if (wave < 4) {
        float* out = XS + rowbase * DCH + ncol;
        for (int i = 0; i < 8; ++i) out[(size_t)(mbase + i) * DCH] = acc[i];
    } else {
        const float bb = cbias[ncol];
        float* out = ACC + rowbase * DCH + ncol;
        for (int i = 0; i < 8; ++i) out[(size_t)(mbase + i) * DCH] = acc[i] + bb;
    }
}

// ---------------------------------------------------------------------------
// Per-node attention scores: ssrc[n] = xs[n].a_src ; sdst[n] = x[n].vdst
// ---------------------------------------------------------------------------
__global__ void scores_kernel(const float* __restrict__ X,
                              const float* __restrict__ XS,
                              const float* __restrict__ asrc,
                              const float* __restrict__ vdst,
                              float* __restrict__ ssrc,
                              float* __restrict__ sdst) {
    int n = blockIdx.x * blockDim.x + threadIdx.x;
    if (n >= N_NODES) return;
    const float4* xr = (const float4*)(X  + (size_t)n * DCH);
    const float4* sr = (const float4*)(XS + (size_t)n * DCH);
    const float4* av = (const float4*)asrc;
    const float4* vv = (const float4*)vdst;
    float s1 = 0.f, s2 = 0.f;
    for (int i = 0; i < 16; ++i) {
        float4 a = sr[i], b = av[i];
        s1 += a.x * b.x + a.y * b.y + a.z * b.z + a.w * b.w;
        float4 c = xr[i], d = vv[i];
        s2 += c.x * d.x + c.y * d.y + c.z * d.z + c.w * d.w;
    }
    ssrc[n] = s1;
    sdst[n] = s2;
}

// ---------------------------------------------------------------------------
// Edge passes
// ---------------------------------------------------------------------------
__device__ __forceinline__ void atomicMaxF(float* addr, float v) {
    if (v >= 0.f) atomicMax((int*)addr, __float_as_int(v));
    else          atomicMin((unsigned int*)addr, __float_as_uint(v));
}

__global__ void edge_alpha_max_kernel(const long long* __restrict__ src,
                                      const long long* __restrict__ dst,
                                      const float* __restrict__ ssrc,
                                      const float* __restrict__ sdst,
                                      float* __restrict__ alpha,
                                      float* __restrict__ m) {
    int e = blockIdx.x * blockDim.x + threadIdx.x;
    if (e >= N_EDGES) return;
    int d = (int)dst[e];
    float s = ssrc[(int)src[e]] + sdst[d];
    float a = s > 0.f ? s : NEG_SLOPE * s;
    alpha[e] = a;
    atomicMaxF(&m[d], a);
}

__global__ void edge_exp_sum_kernel(const long long* __restrict__ dst,
                                    float* __restrict__ alpha,   // in: alpha, out: ex
                                    const float* __restrict__ m,
                                    float* __restrict__ denom) {
    int e = blockIdx.x * blockDim.x + threadIdx.x;
    if (e >= N_EDGES) return;
    int d = (int)dst[e];
    float ex = __expf(alpha[e] - m[d]);
    alpha[e] = ex;
    atomicAdd(&denom[d], ex);
}

// 16 threads per edge; float4 gather of xs[src], 4 f32 atomics into acc[dst]
__global__ void __launch_bounds__(256)
edge_scatter_kernel(const long long* __restrict__ src,
                    const long long* __restrict__ dst,
                    const float* __restrict__ ex,
                    const float* __restrict__ denom,
                    const float* __restrict__ xs,
                    float* __restrict__ acc) {
    int t = blockIdx.x * blockDim.x + threadIdx.x;
    int e = t >> 4;
    if (e >= N_EDGES) return;
    int c4 = (t & 15) * 4;
    int s = (int)src[e];
    int d = (int)dst[e];
    float w = ex[e] / (denom[d] + 1e-16f);
    float4 v = *(const float4*)(xs + (size_t)s * DCH + c4);
    float* o = acc + (size_t)d * DCH + c4;
    atomicAdd(o + 0, v.x * w);
    atomicAdd(o + 1, v.y * w);
    atomicAdd(o + 2, v.z * w);
    atomicAdd(o + 3, v.w * w);
}

__global__ void relu_kernel(float* __restrict__ a, int n) {
    int i = blockIdx.x * blockDim.x + threadIdx.x;
    if (i < n) a[i] = fmaxf(a[i], 0.f);
}

// ---------------------------------------------------------------------------
// Host driver
// ---------------------------------------------------------------------------
extern "C" void kernel_launch(void* const* d_in, const int* in_sizes, int n_in,
                              void* d_out, int out_size, void* d_ws, size_t ws_size,
                              hipStream_t stream) {
    const float*     x    = (const float*)d_in[0];
    const long long* ei   = (const long long*)d_in[1];
    const float* W1_src = (const float*)d_in[2];
    const float* W1_dst = (const float*)d_in[3];
    const float* a1_src = (const float*)d_in[4];
    const float* a1_dst = (const float*)d_in[5];
    const float* b1     = (const float*)d_in[6];
    const float* Wl1    = (const float*)d_in[7];
    const float* bl1    = (const float*)d_in[8];
    const float* W2_src = (const float*)d_in[9];
    const float* W2_dst = (const float*)d_in[10];
    const float* a2_src = (const float*)d_in[11];
    const float* a2_dst = (const float*)d_in[12];
    const float* b2     = (const float*)d_in[13];
    const float* Wl2    = (const float*)d_in[14];
    const float* bl2    = (const float*)d_in[15];

    const long long* src = ei;
    const long long* dst = ei + N_EDGES;

    // Workspace layout (floats)
    float* ws = (float*)d_ws;
    const size_t NF = (size_t)N_NODES * DCH;
    float* xs    = ws;                 // [N,64]
    float* hacc  = xs + NF;            // [N,64] layer-1 accumulator -> h
    float* ssrc  = hacc + NF;          // [N]
    float* sdst  = ssrc + N_NODES;     // [N]
    float* m     = sdst + N_NODES;     // [N]
    float* denom = m + N_NODES;        // [N]
    float* alpha = denom + N_NODES;    // [E] (alpha then ex)
    float* vdst  = alpha + N_EDGES;    // [64]
    float* cbias = vdst + DCH;         // [64]

    float* out = (float*)d_out;

    const int NB_NODE  = N_NODES / 256;            // 256
    const int NB_EDGE  = N_EDGES / 256;            // 4096
    const int NB_SCAT  = (N_EDGES * 16) / 256;     // 65536
    const int NB_GEMM  = N_NODES / 16;             // 4096
    const int NB_RELU  = (int)(NF / 256);          // 16384

    // ---------------- Layer 1 ----------------
    prep_kernel<<<1, 64, 0, stream>>>(W1_dst, a1_dst, b1, bl1, vdst, cbias);
    init_nodes_kernel<<<NB_NODE, 256, 0, stream>>>(m, denom);
    gemm_dual_kernel<<<NB_GEMM, 256, 0, stream>>>(x, W1_src, Wl1, cbias, xs, hacc);
    scores_kernel<<<NB_NODE, 256, 0, stream>>>(x, xs, a1_src, vdst, ssrc, sdst);
    edge_alpha_max_kernel<<<NB_EDGE, 256, 0, stream>>>(src, dst, ssrc, sdst, alpha, m);
    edge_exp_sum_kernel<<<NB_EDGE, 256, 0, stream>>>(dst, alpha, m, denom);
    edge_scatter_kernel<<<NB_SCAT, 256, 0, stream>>>(src, dst, alpha, denom, xs, hacc);
    relu_kernel<<<NB_RELU, 256, 0, stream>>>(hacc, (int)NF);

    // ---------------- Layer 2 ----------------
    prep_kernel<<<1, 64, 0, stream>>>(W2_dst, a2_dst, b2, bl2, vdst, cbias);
    init_nodes_kernel<<<NB_NODE, 256, 0, stream>>>(m, denom);
    gemm_dual_kernel<<<NB_GEMM, 256, 0, stream>>>(hacc, W2_src, Wl2, cbias, xs, out);
    scores_kernel<<<NB_NODE, 256, 0, stream>>>(hacc, xs, a2_src, vdst, ssrc, sdst);
    edge_alpha_max_kernel<<<NB_EDGE, 256, 0, stream>>>(src, dst, ssrc, sdst, alpha, m);
    edge_exp_sum_kernel<<<NB_EDGE, 256, 0, stream>>>(dst, alpha, m, denom);
    edge_scatter_kernel<<<NB_SCAT, 256, 0, stream>>>(src, dst, alpha, denom, xs, out);
}